// KNN_85873576116474
// MI455X (gfx1250) — compile-verified
//
#include <hip/hip_runtime.h>
#include <hip/hip_bf16.h>
#include <stdint.h>

// ---------------- problem constants (from reference) ----------------
#define Q_TOTAL   2048
#define N_TOTAL   100000
#define D_DIM     256
#define K_OUT     16

#define QT        16                     // queries per wave tile (WMMA M)
#define NT        16                     // candidates per tile   (WMMA N)
#define WAVES_WG  4                      // waves per workgroup
#define Q_WG      (QT * WAVES_WG)        // 64 queries per workgroup
#define P_SLICES  50                     // N partitions
#define N_SLICE   (N_TOTAL / P_SLICES)   // 2000
#define TILES     (N_SLICE / NT)         // 125
#define KSTEPS    (D_DIM / 4)            // 64 wmma K-steps of 4
#define BROW      260                    // padded LDS row stride (words) -> conflict-free

typedef float v2f __attribute__((ext_vector_type(2)));
typedef float v8f __attribute__((ext_vector_type(8)));
typedef int   v4i __attribute__((ext_vector_type(4)));

typedef __attribute__((address_space(1))) v4i gv4i;   // global int4
typedef __attribute__((address_space(3))) v4i lv4i;   // LDS int4

#if __has_builtin(__builtin_amdgcn_global_load_async_to_lds_b128) && \
    __has_builtin(__builtin_amdgcn_s_wait_asynccnt)
#define ASYNC_LDS_OK 1
#endif

// 16B global -> LDS async copy, no VGPR data (ASYNCcnt-tracked).
// BYTE_OFF is applied to both global and LDS addresses (IOFFSET semantics).
template <int BYTE_OFF>
static __device__ __forceinline__ void stage16(const float* g, float* l) {
#ifdef ASYNC_LDS_OK
    __builtin_amdgcn_global_load_async_to_lds_b128(
        (gv4i*)(uintptr_t)g,
        (lv4i*)(uint32_t)(uintptr_t)l,   // LDS generic addr carries byte offset in [31:0]
        BYTE_OFF, 0);
#else
    *(float4*)((char*)l + BYTE_OFF) = *(const float4*)((const char*)g + BYTE_OFF);
#endif
}

static __device__ __forceinline__ void stage_wait() {
#ifdef ASYNC_LDS_OK
    __builtin_amdgcn_s_wait_asynccnt(0);
#endif
}

// ---------------- phase 0: y2[n] = ||y_n||^2 ----------------
__global__ void knn_y2_kernel(const float* __restrict__ y, float* __restrict__ y2, int n_total) {
    int n = blockIdx.x * blockDim.x + threadIdx.x;
    if (n >= n_total) return;
    const float4* row = (const float4*)(y + (size_t)n * D_DIM);
    float acc = 0.0f;
#pragma unroll 16
    for (int i = 0; i < D_DIM / 4; ++i) {
        float4 v = row[i];
        acc = fmaf(v.x, v.x, acc);
        acc = fmaf(v.y, v.y, acc);
        acc = fmaf(v.z, v.z, acc);
        acc = fmaf(v.w, v.w, acc);
    }
    y2[n] = acc;
}

// ---------------- phase 1: fused WMMA scores + per-slice top-16 ----------------
// 4 waves / workgroup; each wave owns a distinct 16-query tile; all waves share
// one LDS-staged 16x256 B tile, double buffered via async global->LDS copies.
// grid = (Q/64, P_SLICES), block = 128.
__global__ __launch_bounds__(128) void knn_partial_kernel(
    const float* __restrict__ x, const float* __restrict__ y,
    const float* __restrict__ y2,
    float* __restrict__ outS, int* __restrict__ outI)
{
    __shared__ float sB[2][NT * BROW];        // 2 x 16.6 KB staged B tiles
    __shared__ float sC[WAVES_WG][QT * NT];   // per-wave score tile handoff

    const int tid  = threadIdx.x;             // 0..127
    const int wave = tid >> 5;                // 0..3
    const int lane = tid & 31;                // 0..31
    const int q0   = blockIdx.x * Q_WG + wave * QT;
    const int n0   = blockIdx.y * N_SLICE;

    const int mrow = lane & 15;               // A: M = lane%16
    const int koff = (lane >> 4) << 1;        // A/B: K sub-offset 0 or 2

    // staging role: thread moves 8 x 16B = 128B of the B tile
    const int srow = tid >> 3;                // 0..15
    const int scol = (tid & 7) * 32;          // word offset within row

    // ---- preload A fragments (16x256 query tile, resident across all n-tiles) ----
    v2f a[KSTEPS];
    {
        const float* xrow = x + (size_t)(q0 + mrow) * D_DIM + koff;
#pragma unroll
        for (int k = 0; k < KSTEPS; ++k)
            a[k] = *(const v2f*)(xrow + k * 4);
    }

    // ---- per-lane top-16 list (lanes 0..15 each own one query row) ----
    float sc[K_OUT];
    int   id[K_OUT];
#pragma unroll
    for (int j = 0; j < K_OUT; ++j) { sc[j] = 3.0e38f; id[j] = 0; }

    // ---- prologue: async-stage tile 0 into sB[0] ----
    {
        const float* g = y + (size_t)(n0 + srow) * D_DIM + scol;
        float*       l = &sB[0][srow * BROW + scol];
        stage16<0>(g, l);   stage16<16>(g, l);  stage16<32>(g, l);  stage16<48>(g, l);
        stage16<64>(g, l);  stage16<80>(g, l);  stage16<96>(g, l);  stage16<112>(g, l);
    }

    for (int t = 0; t < TILES; ++t) {
        stage_wait();      // our async copies for tile t have landed in LDS
        __syncthreads();   // all threads' copies visible; prev compute done

        // ---- issue async staging of tile t+1 into the other buffer ----
        if (t + 1 < TILES) {
            const float* g = y + (size_t)(n0 + (t + 1) * NT + srow) * D_DIM + scol;
            float*       l = &sB[(t + 1) & 1][srow * BROW + scol];
            stage16<0>(g, l);   stage16<16>(g, l);  stage16<32>(g, l);  stage16<48>(g, l);
            stage16<64>(g, l);  stage16<80>(g, l);  stage16<96>(g, l);  stage16<112>(g, l);
        }

        // ---- 16x16x256 WMMA block from LDS B fragments ----
        const float* bbase = &sB[t & 1][(lane & 15) * BROW + koff];
        v8f c0 = {};
        v8f c1 = {};
#pragma unroll
        for (int k = 0; k < KSTEPS; k += 2) {
            v2f b0 = *(const v2f*)(bbase + k * 4);
            v2f b1 = *(const v2f*)(bbase + k * 4 + 4);
            c0 = __builtin_amdgcn_wmma_f32_16x16x4_f32(false, a[k],     false, b0,
                                                       (short)0, c0, false, false);
            c1 = __builtin_amdgcn_wmma_f32_16x16x4_f32(false, a[k + 1], false, b1,
                                                       (short)0, c1, false, false);
        }
        v8f dotv = c0 + c1;

        // ---- hand C tile to selection lanes via per-wave LDS region ----
        // C layout: lane L holds col = L%16, rows r + 8*(L/16)
#pragma unroll
        for (int r = 0; r < 8; ++r)
            sC[wave][(r + ((lane >> 4) << 3)) * NT + (lane & 15)] = dotv[r];
        __syncthreads();

        const int nbase = n0 + t * NT;
        if (lane < QT) {
#pragma unroll
            for (int col = 0; col < NT; ++col) {
                const int   n = nbase + col;
                const float s = fmaf(-2.0f, sC[wave][lane * NT + col], y2[n]);
                if (s < sc[K_OUT - 1]) {
                    sc[K_OUT - 1] = s; id[K_OUT - 1] = n;
#pragma unroll
                    for (int j = K_OUT - 1; j > 0; --j) {
                        if (sc[j] < sc[j - 1]) {
                            float tf = sc[j]; sc[j] = sc[j - 1]; sc[j - 1] = tf;
                            int   ti = id[j]; id[j] = id[j - 1]; id[j - 1] = ti;
                        }
                    }
                }
            }
        }
    }

    if (lane < QT) {
        const size_t base = ((size_t)(q0 + lane) * P_SLICES + blockIdx.y) * K_OUT;
#pragma unroll
        for (int j = 0; j < K_OUT; ++j) { outS[base + j] = sc[j]; outI[base + j] = id[j]; }
    }
}

// ---------------- phase 2: merge P_SLICES partial lists per query ----------------
__global__ void knn_merge_kernel(const float* __restrict__ S, const int* __restrict__ I,
                                 int* __restrict__ out)
{
    int q = blockIdx.x * blockDim.x + threadIdx.x;
    if (q >= Q_TOTAL) return;

    float sc[K_OUT];
    int   id[K_OUT];
#pragma unroll
    for (int j = 0; j < K_OUT; ++j) { sc[j] = 3.0e38f; id[j] = 0; }

    const size_t base = (size_t)q * P_SLICES * K_OUT;
    for (int i = 0; i < P_SLICES * K_OUT; ++i) {
        float s = S[base + i];
        if (s < sc[K_OUT - 1]) {
            int n = I[base + i];
            sc[K_OUT - 1] = s; id[K_OUT - 1] = n;
#pragma unroll
            for (int j = K_OUT - 1; j > 0; --j) {
                if (sc[j] < sc[j - 1]) {
                    float tf = sc[j]; sc[j] = sc[j - 1]; sc[j - 1] = tf;
                    int   ti = id[j]; id[j] = id[j - 1]; id[j - 1] = ti;
                }
            }
        }
    }
#pragma unroll
    for (int j = 0; j < K_OUT; ++j) out[q * K_OUT + j] = id[j];
}

// ---------------- launcher ----------------
extern "C" void kernel_launch(void* const* d_in, const int* in_sizes, int n_in,
                              void* d_out, int out_size, void* d_ws, size_t ws_size,
                              hipStream_t stream) {
    const float* x = (const float*)d_in[0];   // [Q, D]
    const float* y = (const float*)d_in[1];   // [N, D]
    // d_in[2] is k == 16, fixed at compile time.
    int* out = (int*)d_out;                   // [Q, K] int32 indices

    // workspace layout: [y2: 400KB][pad to 512KB][partial scores 6.55MB][partial idx 6.55MB]
    char* ws = (char*)d_ws;
    float* y2 = (float*)ws;
    float* pS = (float*)(ws + (512u << 10));
    int*   pI = (int*)  (ws + (512u << 10) + (size_t)Q_TOTAL * P_SLICES * K_OUT * sizeof(float));

    knn_y2_kernel<<<(N_TOTAL + 255) / 256, 256, 0, stream>>>(y, y2, N_TOTAL);

    dim3 grid(Q_TOTAL / Q_WG, P_SLICES);
    knn_partial_kernel<<<grid, WAVES_WG * 32, 0, stream>>>(x, y, y2, pS, pI);

    knn_merge_kernel<<<(Q_TOTAL + 255) / 256, 256, 0, stream>>>(pS, pI, out);
}